// Loss_59923383714465
// MI455X (gfx1250) — compile-verified
//
#include <hip/hip_runtime.h>
#include <hip/hip_bf16.h>

#define VOCAB 32000
#define TM1   1023          // positions per batch row (T-1)
#define NROWS (4 * TM1)     // 4092
#define EPSV  0.1f
#define TPB   256           // 8 wave32s per block

typedef float v2f __attribute__((ext_vector_type(2)));
typedef float v8f __attribute__((ext_vector_type(8)));

// Exact f32 wave32 reduction using V_WMMA_F32_16X16X4_F32.
// Per-lane partial x -> A = {x, 0}:
//   lanes 0-15 : A[m,K0]=x, A[m,K1]=0   (m = lane)
//   lanes16-31 : A[m,K2]=x, A[m,K3]=0   (m = lane-16)
// B = all ones  =>  D[m,n] = rowsum_A(m) = x[m] + x[m+16]  (independent of n)
// D layout: lanes 0-15 hold M=0..7 in d[0..7]; lanes 16-31 hold M=8..15.
// Sum d[0..7] per lane, then xor-shuffle across half-waves -> total in all lanes.
__device__ __forceinline__ float wave_reduce_wmma(float x) {
    v2f a; a.x = x;   a.y = 0.0f;
    v2f b; b.x = 1.0f; b.y = 1.0f;
    v8f c = {};
    v8f d = __builtin_amdgcn_wmma_f32_16x16x4_f32(
        /*neg_a=*/false, a, /*neg_b=*/false, b,
        /*c_mod=*/(short)0, c, /*reuse_a=*/false, /*reuse_b=*/false);
    float s = ((d[0] + d[1]) + (d[2] + d[3])) + ((d[4] + d[5]) + (d[6] + d[7]));
    s += __shfl_xor(s, 16, 32);
    return s;
}

// ---------------- Kernel A: scalars H = sum(h), Asum = sum a_v ----------------
__global__ void __launch_bounds__(1024)
prep_scalars(const float* __restrict__ h, float* __restrict__ scal) {
    __shared__ float red[1024];
    __shared__ float Hs;
    const int tid = threadIdx.x;

    float s = 0.0f;
    for (int v = tid; v < VOCAB; v += 1024) s += h[v];
    red[tid] = s;
    __syncthreads();
    for (int off = 512; off > 0; off >>= 1) {
        if (tid < off) red[tid] += red[tid + off];
        __syncthreads();
    }
    if (tid == 0) Hs = red[0];
    __syncthreads();

    const float H = Hs;
    const float cEH = EPSV / H;           // eps * u_v = cEH * h_v
    float acc = 0.0f;
    for (int v = tid; v < VOCAB; v += 1024) {
        float w = cEH * h[v];             // w > 0 by construction
        acc += w * __logf(w);
    }
    red[tid] = acc;
    __syncthreads();
    for (int off = 512; off > 0; off >>= 1) {
        if (tid < off) red[tid] += red[tid + off];
        __syncthreads();
    }
    if (tid == 0) { scal[0] = H; scal[1] = red[0]; }
}

// ------------- Kernel B: one block per (b,t) row; S' = sum h_v*log p_v -------------
__global__ void __launch_bounds__(TPB)
row_kl(const int*   __restrict__ dec_input,
       const float* __restrict__ dec_output,
       const float* __restrict__ h,
       const float* __restrict__ scal,
       float*       __restrict__ rkl) {
    const int r = blockIdx.x;             // 0..NROWS-1
    const int b = r / TM1;
    const int t = r - b * TM1;
    const float* __restrict__ prow = dec_output + (size_t)(b * 1024 + t) * VOCAB;

    float acc = 0.0f;
    #pragma unroll 4
    for (int v = threadIdx.x * 4; v < VOCAB; v += TPB * 4) {
        const float4 p4 = *(const float4*)(prow + v);   // global_load_b128
        const float4 h4 = *(const float4*)(h + v);      // L2/WGP$ resident
        acc += h4.x * __logf(p4.x);
        acc += h4.y * __logf(p4.y);
        acc += h4.z * __logf(p4.z);
        acc += h4.w * __logf(p4.w);
    }

    // exact f32 wave reduction on the matrix unit (EXEC full: loop reconverged)
    const float wsum = wave_reduce_wmma(acc);

    __shared__ float lds[TPB / 32];
    const int lane = threadIdx.x & 31;
    const int wv   = threadIdx.x >> 5;
    if (lane == 0) lds[wv] = wsum;
    __syncthreads();

    if (threadIdx.x == 0) {
        float S = 0.0f;
        #pragma unroll
        for (int i = 0; i < TPB / 32; ++i) S += lds[i];   // fixed order

        const float H    = scal[0];
        const float Asum = scal[1];
        const float cEH  = EPSV / H;
        const int   l    = dec_input[b * 1024 + t + 1];   // label
        const float w_l  = cEH * h[l];                    // eps*u_l
        const float p_l  = prow[l];
        const float q_l  = w_l + (1.0f - EPSV);
        const float a_l  = w_l * __logf(w_l);
        const float kl   = Asum - a_l + q_l * __logf(q_l)
                         - cEH * S - (1.0f - EPSV) * __logf(p_l);
        rkl[r] = (l != 0) ? kl : 0.0f;                    // PAD == 0 mask
    }
}

// ------------- Kernel C: deterministic mean over NROWS -------------
__global__ void __launch_bounds__(1024)
final_mean(const float* __restrict__ rkl, float* __restrict__ out) {
    __shared__ float red[1024];
    const int tid = threadIdx.x;
    float s = 0.0f;
    for (int r = tid; r < NROWS; r += 1024) s += rkl[r];  // fixed order
    red[tid] = s;
    __syncthreads();
    for (int off = 512; off > 0; off >>= 1) {
        if (tid < off) red[tid] += red[tid + off];
        __syncthreads();
    }
    if (tid == 0) out[0] = red[0] * (1.0f / (float)NROWS);
}

extern "C" void kernel_launch(void* const* d_in, const int* in_sizes, int n_in,
                              void* d_out, int out_size, void* d_ws, size_t ws_size,
                              hipStream_t stream) {
    (void)in_sizes; (void)n_in; (void)out_size; (void)ws_size;
    const int*   dec_input  = (const int*)d_in[0];    // (4,1024) int32
    const float* dec_output = (const float*)d_in[1];  // (4,1024,32000) f32
    const float* token_hist = (const float*)d_in[2];  // (32000,) f32
    float* out = (float*)d_out;

    float* ws_f  = (float*)d_ws;
    float* rkl   = ws_f;            // NROWS floats
    float* scal  = ws_f + 4096;     // [H, Asum]

    prep_scalars<<<1, 1024, 0, stream>>>(token_hist, scal);
    row_kl<<<NROWS, TPB, 0, stream>>>(dec_input, dec_output, token_hist, scal, rkl);
    final_mean<<<1, 1024, 0, stream>>>(rkl, out);
}